// Net_191120_7670811590820
// MI455X (gfx1250) — compile-verified
//
#include <hip/hip_runtime.h>
#include <math.h>

typedef __attribute__((ext_vector_type(2))) float v2f;
typedef __attribute__((ext_vector_type(8))) float v8f;
typedef __attribute__((ext_vector_type(4))) int   v4i;

#if defined(__has_builtin)
#if __has_builtin(__builtin_amdgcn_tensor_load_to_lds) && __has_builtin(__builtin_amdgcn_s_wait_tensorcnt)
#define HAVE_TDM 1
#endif
#endif

#ifdef HAVE_TDM
typedef __attribute__((ext_vector_type(4))) unsigned int v4u;
typedef __attribute__((ext_vector_type(8))) int          v8i;
#endif

// ---------------------------------------------------------------------------
// Collapsed GCN:  out = log_softmax( meanpool( A*(A*x*W1 + b1)*W2 + b2 ) )
//               = log_softmax( mean_z * (W1W2) + mean_s * (b1'W2) + b2 )
// with z = A^2 x, s = A*1 (A = D^-1/2 (Adj+I) D^-1/2), all scalar per node.
// Node arrays are L2-resident (~3.2MB << 192MB); edge-index streams are
// loaded B128 + non-temporal so they do not evict the hot node data.
// ---------------------------------------------------------------------------

__global__ void k_init_deg(float* __restrict__ deg, int n) {
  int i = blockIdx.x * blockDim.x + threadIdx.x;
  if (i < n) deg[i] = 1.0f;  // self-loop contributes 1 to every degree
}

// ---- degree count: 4 edges / thread, NT B128 index loads
__global__ void k_count4(const v4i* __restrict__ dst4, float* __restrict__ deg, int e4) {
  int i = blockIdx.x * blockDim.x + threadIdx.x;
  if (i < e4) {
    v4i d = __builtin_nontemporal_load(dst4 + i);
#pragma unroll
    for (int k = 0; k < 4; ++k) atomicAdd(&deg[d[k]], 1.0f);
  }
}
__global__ void k_count(const int* __restrict__ dst, float* __restrict__ deg, int e) {
  int i = blockIdx.x * blockDim.x + threadIdx.x;
  if (i < e) atomicAdd(&deg[dst[i]], 1.0f);
}

// deg -> dinv (in place); u = dinv*x; ytmp init = u (self loop); stmp init = dinv
__global__ void k_prep1(const float* __restrict__ x, float* __restrict__ dinv,
                        float* __restrict__ u, float* __restrict__ ytmp,
                        float* __restrict__ stmp, int n) {
  int i = blockIdx.x * blockDim.x + threadIdx.x;
  if (i < n) {
    float d = rsqrtf(dinv[i]);   // dinv[] currently holds deg (>= 1)
    dinv[i] = d;
    float ui = d * x[i];
    u[i] = ui;
    ytmp[i] = ui;                // self-loop term (pre dinv[dst] scaling)
    stmp[i] = d;                 // self-loop term of s
  }
}

// ---- edge sweep 1: ytmp[dst] += u[src]; stmp[dst] += dinv[src]
__global__ void k_edge1v(const v4i* __restrict__ src4, const v4i* __restrict__ dst4,
                         const float* __restrict__ u, const float* __restrict__ dinv,
                         float* __restrict__ ytmp, float* __restrict__ stmp, int e4) {
  int i = blockIdx.x * blockDim.x + threadIdx.x;
  if (i < e4) {
    v4i s = __builtin_nontemporal_load(src4 + i);
    v4i d = __builtin_nontemporal_load(dst4 + i);
#pragma unroll
    for (int k = 0; k < 4; ++k) {
      atomicAdd(&ytmp[d[k]], u[s[k]]);
      atomicAdd(&stmp[d[k]], dinv[s[k]]);
    }
  }
}
__global__ void k_edge1(const int* __restrict__ src, const int* __restrict__ dst,
                        const float* __restrict__ u, const float* __restrict__ dinv,
                        float* __restrict__ ytmp, float* __restrict__ stmp, int e) {
  int i = blockIdx.x * blockDim.x + threadIdx.x;
  if (i < e) {
    int s = src[i], d = dst[i];
    atomicAdd(&ytmp[d], u[s]);
    atomicAdd(&stmp[d], dinv[s]);
  }
}

// y = dinv*ytmp; v = dinv*y; ztmp(in a2) init = v; s(in stmp) = dinv*stmp
__global__ void k_prep2(const float* __restrict__ dinv, float* __restrict__ a2,
                        float* __restrict__ v, float* __restrict__ stmp, int n) {
  int i = blockIdx.x * blockDim.x + threadIdx.x;
  if (i < n) {
    float d  = dinv[i];
    float y  = d * a2[i];
    float vi = d * y;
    v[i]  = vi;
    a2[i] = vi;                  // becomes ztmp self-loop init
    stmp[i] = d * stmp[i];       // final s[i]
  }
}

// ---- edge sweep 2: ztmp[dst] += v[src]
__global__ void k_edge2v(const v4i* __restrict__ src4, const v4i* __restrict__ dst4,
                         const float* __restrict__ v, float* __restrict__ ztmp, int e4) {
  int i = blockIdx.x * blockDim.x + threadIdx.x;
  if (i < e4) {
    v4i s = __builtin_nontemporal_load(src4 + i);
    v4i d = __builtin_nontemporal_load(dst4 + i);
#pragma unroll
    for (int k = 0; k < 4; ++k) atomicAdd(&ztmp[d[k]], v[s[k]]);
  }
}
__global__ void k_edge2(const int* __restrict__ src, const int* __restrict__ dst,
                        const float* __restrict__ v, float* __restrict__ ztmp, int e) {
  int i = blockIdx.x * blockDim.x + threadIdx.x;
  if (i < e) atomicAdd(&ztmp[dst[i]], v[src[i]]);
}

__global__ void k_zero_pool(float* __restrict__ gz, float* __restrict__ gs,
                            float* __restrict__ gc) {
  int i = threadIdx.x;
  if (i < 128) { gz[i] = 0.0f; gs[i] = 0.0f; gc[i] = 0.0f; }
}

__global__ void k_pool(const float* __restrict__ dinv, const float* __restrict__ ztmp,
                       const float* __restrict__ s, const int* __restrict__ batch,
                       float* __restrict__ gz, float* __restrict__ gs,
                       float* __restrict__ gc, int n) {
  __shared__ float lz[128], ls[128], lc[128];
  int t = threadIdx.x;
  if (t < 128) { lz[t] = 0.0f; ls[t] = 0.0f; lc[t] = 0.0f; }
  __syncthreads();
  int i = blockIdx.x * blockDim.x + t;
  if (i < n) {
    int g = batch[i] & 127;
    float z = dinv[i] * ztmp[i];
    atomicAdd(&lz[g], z);
    atomicAdd(&ls[g], s[i]);
    atomicAdd(&lc[g], 1.0f);
  }
  __syncthreads();
  if (t < 128 && lc[t] != 0.0f) {
    atomicAdd(&gz[t], lz[t]);
    atomicAdd(&gs[t], ls[t]);
    atomicAdd(&gc[t], lc[t]);
  }
}

// Single block, 128 threads. Wave 0 (full EXEC) runs two WMMA stages:
//  Stage 1: [W1;b1] (2x32) x W2 (32x8) -> w12 (row0), bw (row1): 8x v_wmma f32 16x16x4
//  Stage 2: M[128x4] x [w12;bw;b2;0] -> pooled[128x8]: 8 tiles of same op
// W2 is staged into a zero-padded [32][16] LDS tile by the Tensor Data Mover
// (2D tile, 8-DW rows, LDS pad_interval=8DW / pad_amount=8DW -> 16-DW pitch).
__global__ void k_final(const float* __restrict__ W1, const float* __restrict__ b1,
                        const float* __restrict__ W2, const float* __restrict__ b2,
                        const float* __restrict__ gz, const float* __restrict__ gs,
                        const float* __restrict__ gc, float* __restrict__ out) {
  __shared__ float sA[16 * 32];    // A fragment source: row0=W1, row1=b1, rest 0
  __shared__ float sW2[32 * 16];   // W2 padded to 16 cols
  __shared__ float sb2[16];        // b2 padded to 16
  __shared__ float spool[128 * 8];
  int tid = threadIdx.x;

  // disjoint-writer LDS init (no pre-barrier races)
  for (int i = 64 + tid; i < 16 * 32; i += 128) sA[i] = 0.f;     // rows 2..15 = 0
  if (tid < 64) sA[tid] = (tid < 32) ? W1[tid] : b1[tid - 32];   // rows 0,1
  for (int r = tid; r < 32; r += 128) {
#pragma unroll
    for (int c = 8; c < 16; ++c) sW2[r * 16 + c] = 0.f;          // pad cols = 0
  }
  if (tid < 16) sb2[tid] = (tid < 8) ? b2[tid] : 0.f;
  __syncthreads();

  if (tid < 32) {
#ifdef HAVE_TDM
    // ---- TDM: W2[32][8] (global, row stride 8) -> sW2[32][16] (LDS, padded)
    unsigned long long ga = (unsigned long long)(uintptr_t)W2;
    unsigned ldsoff = (unsigned)(uintptr_t)(&sW2[0]);            // LDS byte offset
    v4u g0;
    g0[0] = 1u;                                                  // count=1, user D#
    g0[1] = ldsoff;                                              // lds_addr
    g0[2] = (unsigned)ga;                                        // global_addr[31:0]
    g0[3] = (unsigned)((ga >> 32) & 0x1FFFFFFu) | (2u << 30);    // addr[56:32], type=2
    v8i g1;
    g1[0] = (int)((2u << 16) |      // data_size = 4B
                  (1u << 20) |      // pad_enable
                  (2u << 22) |      // pad_interval: every 8 DWORDs
                  (7u << 25));      // pad_amount: 8 DWORDs
    g1[1] = (int)(8u << 16);        // tensor_dim0 = 8
    g1[2] = (int)(32u << 16);       // tensor_dim1 = 32
    g1[3] = (int)(8u << 16);        // tile_dim0 = 8
    g1[4] = 32;                     // tile_dim1 = 32
    g1[5] = 8;                      // tensor_dim0_stride = 8
    g1[6] = 0;
    g1[7] = 0;
    v4i gnull4  = {0, 0, 0, 0};
    v8i gnull8  = {0, 0, 0, 0, 0, 0, 0, 0};
    __builtin_amdgcn_tensor_load_to_lds(g0, g1, gnull4, gnull4, gnull8, 0);
    __builtin_amdgcn_s_wait_tensorcnt((short)0);
#else
    for (int i = tid; i < 256; i += 32) sW2[(i >> 3) * 16 + (i & 7)] = W2[i];
#endif

    int  col = tid & 15;
    bool hi  = tid >= 16;            // lanes 16-31 hold K+2 (A) / rows 2-3 (B)

    // ---- Stage 1: acc = [W1; b1; 0...] (16x32) x W2pad (32x16)
    v8f acc = {0.f, 0.f, 0.f, 0.f, 0.f, 0.f, 0.f, 0.f};
#pragma unroll
    for (int kk = 0; kk < 8; ++kk) {
      int kb = kk * 4 + (hi ? 2 : 0);
      v2f a, b;
      a.x = sA[col * 32 + kb];
      a.y = sA[col * 32 + kb + 1];
      b.x = sW2[kb * 16 + col];
      b.y = sW2[(kb + 1) * 16 + col];
      acc = __builtin_amdgcn_wmma_f32_16x16x4_f32(
          false, a, false, b, (short)0, acc, false, false);
    }
    // D layout: lanes<16, acc[0] = w12[col], acc[1] = bw[col] (cols>=8 are 0).

    // ---- Stage 2: pooled = [mz, ms, 1, 0] x [w12; bw; b2; 0]
    v2f bb;
    bb.x = hi ? sb2[col] : acc[0];   // rows 0 / 2 of C
    bb.y = hi ? 0.f      : acc[1];   // rows 1 / 3 of C
#pragma unroll
    for (int t2 = 0; t2 < 8; ++t2) {
      int g = t2 * 16 + col;
      float c  = fmaxf(gc[g], 1.0f);
      float mz = gz[g] / c;
      float ms = gs[g] / c;
      v2f aa;
      aa.x = hi ? 1.0f : mz;         // K=2 selects b2 row / K=0 mean_z
      aa.y = hi ? 0.0f : ms;         // K=3 zero        / K=1 mean_s
      v8f d = {0.f, 0.f, 0.f, 0.f, 0.f, 0.f, 0.f, 0.f};
      d = __builtin_amdgcn_wmma_f32_16x16x4_f32(
          false, aa, false, bb, (short)0, d, false, false);
#pragma unroll
      for (int r = 0; r < 8; ++r) {
        int row = t2 * 16 + r + (hi ? 8 : 0);
        if (col < 8) spool[row * 8 + col] = d[r];
      }
    }
  }
  __syncthreads();

  // ---- log_softmax, one graph per thread (blockDim.x == 128)
  int g = tid;
  float vals[8];
  float mx = -3.402823466e38f;
#pragma unroll
  for (int k = 0; k < 8; ++k) { vals[k] = spool[g * 8 + k]; mx = fmaxf(mx, vals[k]); }
  float se = 0.f;
#pragma unroll
  for (int k = 0; k < 8; ++k) se += expf(vals[k] - mx);
  float lse = mx + logf(se);
#pragma unroll
  for (int k = 0; k < 8; ++k) out[g * 8 + k] = vals[k] - lse;
}

extern "C" void kernel_launch(void* const* d_in, const int* in_sizes, int n_in,
                              void* d_out, int out_size, void* d_ws, size_t ws_size,
                              hipStream_t stream) {
  const float* x     = (const float*)d_in[0];
  const int*   ei    = (const int*)d_in[1];   // [2, E]: row0 = src, row1 = dst
  const int*   batch = (const int*)d_in[2];
  const float* W1    = (const float*)d_in[3];
  const float* b1    = (const float*)d_in[4];
  const float* W2    = (const float*)d_in[5];
  const float* b2    = (const float*)d_in[6];
  float*       out   = (float*)d_out;

  int N = in_sizes[0];
  int E = in_sizes[1] / 2;

  float* ws   = (float*)d_ws;
  float* dinv = ws;                   // deg -> dinv
  float* u    = ws + (size_t)N;       // u, then v
  float* a2   = ws + (size_t)2 * N;   // ytmp, then ztmp
  float* stmp = ws + (size_t)3 * N;   // stmp, then s
  float* gz   = ws + (size_t)4 * N;
  float* gs   = gz + 128;
  float* gc   = gs + 128;

  const int* src = ei;
  const int* dst = ei + E;

  int nb = (N + 255) / 256;
  int eb = (E + 255) / 256;

  k_init_deg<<<nb, 256, 0, stream>>>(dinv, N);

  if ((E & 3) == 0 && E > 0) {
    int e4  = E >> 2;
    int eb4 = (e4 + 255) / 256;
    k_count4<<<eb4, 256, 0, stream>>>((const v4i*)dst, dinv, e4);
    k_prep1 <<<nb, 256, 0, stream>>>(x, dinv, u, a2, stmp, N);
    k_edge1v<<<eb4, 256, 0, stream>>>((const v4i*)src, (const v4i*)dst, u, dinv, a2, stmp, e4);
    k_prep2 <<<nb, 256, 0, stream>>>(dinv, a2, u, stmp, N);
    k_edge2v<<<eb4, 256, 0, stream>>>((const v4i*)src, (const v4i*)dst, u, a2, e4);
  } else {
    k_count<<<eb, 256, 0, stream>>>(dst, dinv, E);
    k_prep1<<<nb, 256, 0, stream>>>(x, dinv, u, a2, stmp, N);
    k_edge1<<<eb, 256, 0, stream>>>(src, dst, u, dinv, a2, stmp, E);
    k_prep2<<<nb, 256, 0, stream>>>(dinv, a2, u, stmp, N);
    k_edge2<<<eb, 256, 0, stream>>>(src, dst, u, a2, E);
  }

  k_zero_pool<<<1, 128, 0, stream>>>(gz, gs, gc);
  k_pool     <<<nb, 256, 0, stream>>>(dinv, a2, stmp, batch, gz, gs, gc, N);
  k_final    <<<1, 128, 0, stream>>>(W1, b1, W2, b2, gz, gs, gc, out);
}